// PEwithPeak_15934328668242
// MI455X (gfx1250) — compile-verified
//
#include <hip/hip_runtime.h>
#include <math.h>

typedef __attribute__((ext_vector_type(4))) float v4f;

#define S_LEN 256
#define B_SZ  2048
#define DIM   256
#define NPK   8
#define BB    32   // batches per block (BB*NPK == 256 == blockDim)

__global__ __launch_bounds__(256) void pe_peak_kernel(
    const float* __restrict__ x,
    const int*   __restrict__ pp,       // peak_positions [B, 8]
    const float* __restrict__ table,    // [256, 256]
    float*       __restrict__ out)
{
    __shared__ float s_pe[DIM];
    __shared__ float s_tab[DIM];
    __shared__ int   s_flag[BB];

    const int tid = threadIdx.x;
    const int s   = blockIdx.x;               // sequence position, 0..255
    const int b0  = blockIdx.y * BB;          // first batch of this block

    // ---- CDNA5 async global->LDS copy of table[s] (1 KB, 64 lanes x 16B) ----
    if (tid < 64) {
        unsigned lds_addr = (unsigned)(size_t)(&s_tab[0]) + (unsigned)tid * 16u;
        unsigned voff     = (unsigned)tid * 16u;
        unsigned long long sbase =
            (unsigned long long)(const void*)(table + (size_t)s * DIM);
        asm volatile("global_load_async_to_lds_b128 %0, %1, %2"
                     :: "v"(lds_addr), "v"(voff), "s"(sbase)
                     : "memory");
    }

    // ---- overlapped with the copy: PE row + peak flags ----
    if (tid < BB) s_flag[tid] = 0;
    {
        // div_term = exp(-(2j) * ln(1000)/DIM), pe[2j]=sin, pe[2j+1]=cos
        const float kk = -6.907755278982137f / (float)DIM; // -ln(1000)/DIM
        int   j   = tid >> 1;
        float ang = (float)s * expf((float)(2 * j) * kk);
        s_pe[tid] = (tid & 1) ? cosf(ang) : sinf(ang);
    }
    // one peak-position load per thread: batch b0 + tid/8, peak tid%8
    int mypos = pp[(size_t)b0 * NPK + (size_t)tid];
    __syncthreads();                       // s_flag init visible
    if (mypos == s) s_flag[tid >> 3] = 1;  // benign race: all write 1

    if (tid < 64) {
        asm volatile("s_wait_asynccnt 0" ::: "memory");  // table row landed
    }
    __syncthreads();                       // pe, flags, table visible to all

    // ---- streaming phase: 32 rows x 256 floats, float4 per lane ----
    const int d4   = tid & 63;             // float4 index along D
    const int rsub = tid >> 6;             // 0..3 row sub-index
    v4f pe4  = *(const v4f*)(&s_pe [d4 * 4]);
    v4f tab4 = *(const v4f*)(&s_tab[d4 * 4]);
    v4f both = pe4 + tab4;

    size_t base = ((size_t)s * B_SZ + (size_t)b0) * DIM + (size_t)d4 * 4;
    #pragma unroll
    for (int it = 0; it < BB / 4; ++it) {
        int    r   = it * 4 + rsub;        // local batch row 0..31
        size_t off = base + (size_t)r * DIM;
        v4f xv  = __builtin_nontemporal_load((const v4f*)(x + off));
        v4f add = s_flag[r] ? both : pe4;
        v4f o   = xv + add;
        __builtin_nontemporal_store(o, (v4f*)(out + off));
    }
}

extern "C" void kernel_launch(void* const* d_in, const int* in_sizes, int n_in,
                              void* d_out, int out_size, void* d_ws, size_t ws_size,
                              hipStream_t stream) {
    (void)in_sizes; (void)n_in; (void)out_size; (void)d_ws; (void)ws_size;
    const float* x     = (const float*)d_in[0];
    const int*   pp    = (const int*)  d_in[1];   // int32 per harness convention
    const float* table = (const float*)d_in[2];
    float*       out   = (float*)d_out;

    dim3 grid(S_LEN, B_SZ / BB, 1);   // 256 x 64 blocks
    dim3 block(256, 1, 1);            // 8 wave32 waves
    pe_peak_kernel<<<grid, block, 0, stream>>>(x, pp, table, out);
}